// WindowedLSTM_75479755259950
// MI455X (gfx1250) — compile-verified
//
#include <hip/hip_runtime.h>
#include <hip/hip_bf16.h>
#include <math.h>

// ---------------- problem constants ----------------
constexpr int kSEQ  = 8192;
constexpr int kIN   = 180;
constexpr int kHID  = 64;
constexpr int kOUT  = 10;
constexpr int kG    = 4 * kHID;    // 256 gates per layer
constexpr int kHPM  = 318;         // OUT + 2*HID + IN

typedef __attribute__((ext_vector_type(2))) float v2f;
typedef __attribute__((ext_vector_type(8))) float v8f;

__device__ __forceinline__ float sigmoidf(float x) {
    return 1.0f / (1.0f + __expf(-x));
}

// =====================================================================
// Kernel 1: pre0[s][g] = b_ih0[g] + b_hh0[g] + sum_k input[s][k]*W_ih0[g][k]
// fp32 WMMA 16x16x4, one 16x16 output tile per wave.
// =====================================================================
__global__ __launch_bounds__(256) void xproj_wmma(
    const float* __restrict__ x,       // [kSEQ][kIN]
    const float* __restrict__ Wih0,    // [kG][kIN]
    const float* __restrict__ bih0,
    const float* __restrict__ bhh0,
    float* __restrict__ pre0)          // [kSEQ][kG]
{
    const int wave = threadIdx.x >> 5;
    const int lane = threadIdx.x & 31;
    const int tile = blockIdx.x * (blockDim.x >> 5) + wave;   // 8192 tiles
    const int tM = tile >> 4;          // 512 tiles along SEQ
    const int tN = tile & 15;          // 16 tiles along gates
    const int s0 = tM * 16;
    const int n0 = tN * 16;
    const int half = lane >> 4;        // 0..1
    const int mr   = lane & 15;        // 0..15

    v8f c = {};

#if __has_builtin(__builtin_amdgcn_wmma_f32_16x16x4_f32)
    // A: 16x4 fp32  (lane m=mr, VGPR j holds K = 2*half + j)
    // B: 4x16 fp32  (lane n=mr, VGPR j holds K = 2*half + j), B[k][n] = Wih0[n][k]
    for (int k0 = 0; k0 < kIN; k0 += 4) {
        const int ka = k0 + 2 * half;
        v2f a, b;
        a.x = x[(s0 + mr) * kIN + ka];
        a.y = x[(s0 + mr) * kIN + ka + 1];
        b.x = Wih0[(n0 + mr) * kIN + ka];
        b.y = Wih0[(n0 + mr) * kIN + ka + 1];
        c = __builtin_amdgcn_wmma_f32_16x16x4_f32(
                /*neg_a=*/false, a, /*neg_b=*/false, b,
                /*c_mod=*/(short)0, c, /*reuse_a=*/false, /*reuse_b=*/false);
    }
#else
    // VALU fallback producing the identical C layout
    #pragma unroll
    for (int r = 0; r < 8; ++r) {
        const int m = r + 8 * half;
        const int n = mr;
        float acc = 0.0f;
        for (int k = 0; k < kIN; ++k)
            acc += x[(s0 + m) * kIN + k] * Wih0[(n0 + n) * kIN + k];
        c[r] = acc;
    }
#endif

    // C layout: VGPR r -> row m = r + 8*half, col n = lane&15
    #pragma unroll
    for (int r = 0; r < 8; ++r) {
        const int m = r + 8 * half;
        const int g = n0 + mr;
        pre0[(s0 + m) * kG + g] = c[r] + bih0[g] + bhh0[g];
    }
}

// =====================================================================
// Kernel 2: sequential 2-layer LSTM scan. ONE workgroup (256 threads,
// 8 wave32) on one WGP. All recurrent weights live in LDS (~198 KB of
// the 320 KB WGP LDS), transposed + float4-packed for conflict-free
// ds_load_b128 across lanes.
//   packed layout: wq[j4*1024 + g*4 + e] = W[g][4*j4 + e]
// =====================================================================
// LDS float offsets
constexpr int OFF_WQ0  = 0;                  // 16384
constexpr int OFF_WQ1  = OFF_WQ0 + 16384;    // 16384 (W_ih1)
constexpr int OFF_WQH1 = OFF_WQ1 + 16384;    // 16384 (W_hh1)
constexpr int OFF_G0   = OFF_WQH1 + 16384;   // 256
constexpr int OFF_G1   = OFF_G0 + 256;       // 256
constexpr int OFF_BS1  = OFF_G1 + 256;       // 256
constexpr int OFF_H0   = OFF_BS1 + 256;      // 64
constexpr int OFF_H1   = OFF_H0 + 64;        // 64
constexpr int OFF_WLIN = OFF_H1 + 64;        // 640
constexpr int OFF_BLIN = OFF_WLIN + 640;     // 16 (padded)
constexpr int LDS_FLOATS = OFF_BLIN + 16;    // 50704 floats = 202816 B

__global__ __launch_bounds__(256) void lstm_seq(
    const float* __restrict__ pre0,    // [kSEQ][kG]
    const float* __restrict__ Whh0,    // [kG][kHID]
    const float* __restrict__ Wih1,    // [kG][kHID]
    const float* __restrict__ Whh1,    // [kG][kHID]
    const float* __restrict__ bih1,
    const float* __restrict__ bhh1,
    const float* __restrict__ Wlin,    // [kOUT][kHID]
    const float* __restrict__ blin,
    float* __restrict__ outY,          // [kSEQ][kOUT]  (d_out region 0)
    float* __restrict__ oldh,          // [kSEQ][kHID]  (ws)
    float* __restrict__ minv,          // [kHID]
    float* __restrict__ maxv)          // [kHID]
{
    extern __shared__ __attribute__((aligned(16))) float sm[];
    const int tid = threadIdx.x;

    // ---- cooperative load of weights into packed-transposed LDS ----
    for (int idx = tid; idx < 16384; idx += 256) {
        const int j4 = idx >> 10;          // 0..15
        const int rem = idx & 1023;
        const int g = rem >> 2;            // 0..255
        const int e = rem & 3;             // 0..3
        const int j = 4 * j4 + e;
        sm[OFF_WQ0  + idx] = Whh0[g * kHID + j];
        sm[OFF_WQ1  + idx] = Wih1[g * kHID + j];
        sm[OFF_WQH1 + idx] = Whh1[g * kHID + j];
    }
    sm[OFF_BS1 + tid] = bih1[tid] + bhh1[tid];
    for (int idx = tid; idx < kOUT * kHID; idx += 256)
        sm[OFF_WLIN + idx] = Wlin[idx];
    if (tid < kOUT) sm[OFF_BLIN + tid] = blin[tid];
    if (tid < kHID) { sm[OFF_H0 + tid] = 0.0f; sm[OFF_H1 + tid] = 0.0f; }
    __syncthreads();

    float c0r = 0.0f, c1r = 0.0f;               // cell states (lanes 0..63)
    float mn = 3.4e38f, mx = -3.4e38f;          // running min/max of old h0

    const float4* wq0  = (const float4*)(sm + OFF_WQ0);
    const float4* wq1  = (const float4*)(sm + OFF_WQ1);
    const float4* wqh1 = (const float4*)(sm + OFF_WQH1);
    const float4* h0v4 = (const float4*)(sm + OFF_H0);
    const float4* h1v4 = (const float4*)(sm + OFF_H1);

    for (int t = 0; t < kSEQ; ++t) {
        // ---- layer-0 gates: g0[tid] = pre0 + h0 . Whh0[tid] ----
        float acc = pre0[t * kG + tid];
        #pragma unroll
        for (int j4 = 0; j4 < 16; ++j4) {
            const float4 w = wq0[j4 * 256 + tid];   // conflict-free b128
            const float4 h = h0v4[j4];              // broadcast
            acc += w.x * h.x + w.y * h.y + w.z * h.z + w.w * h.w;
        }
        sm[OFF_G0 + tid] = acc;
        __syncthreads();

        // ---- layer-0 cell update (lanes 0..63) ----
        if (tid < kHID) {
            const float hOld = sm[OFF_H0 + tid];
            oldh[t * kHID + tid] = hOld;
            mn = fminf(mn, hOld);
            mx = fmaxf(mx, hOld);
            const float ig = sigmoidf(sm[OFF_G0 + tid]);
            const float fg = sigmoidf(sm[OFF_G0 + kHID + tid]);
            const float gg = tanhf(sm[OFF_G0 + 2 * kHID + tid]);
            const float og = sigmoidf(sm[OFF_G0 + 3 * kHID + tid]);
            c0r = fg * c0r + ig * gg;
            sm[OFF_H0 + tid] = og * tanhf(c0r);
        }
        __syncthreads();

        // ---- layer-1 gates: g1[tid] = bsum1 + h0new.Wih1 + h1.Whh1 ----
        float acc1 = sm[OFF_BS1 + tid];
        #pragma unroll
        for (int j4 = 0; j4 < 16; ++j4) {
            const float4 wi = wq1 [j4 * 256 + tid];
            const float4 wh = wqh1[j4 * 256 + tid];
            const float4 h0 = h0v4[j4];
            const float4 h1 = h1v4[j4];
            acc1 += wi.x * h0.x + wi.y * h0.y + wi.z * h0.z + wi.w * h0.w;
            acc1 += wh.x * h1.x + wh.y * h1.y + wh.z * h1.z + wh.w * h1.w;
        }
        sm[OFF_G1 + tid] = acc1;
        __syncthreads();

        // ---- layer-1 cell update (lanes 0..63) ----
        if (tid < kHID) {
            const float ig = sigmoidf(sm[OFF_G1 + tid]);
            const float fg = sigmoidf(sm[OFF_G1 + kHID + tid]);
            const float gg = tanhf(sm[OFF_G1 + 2 * kHID + tid]);
            const float og = sigmoidf(sm[OFF_G1 + 3 * kHID + tid]);
            c1r = fg * c1r + ig * gg;
            sm[OFF_H1 + tid] = og * tanhf(c1r);
        }
        __syncthreads();

        // ---- linear head (threads 0..9) ----
        if (tid < kOUT) {
            float o = sm[OFF_BLIN + tid];
            #pragma unroll 8
            for (int j = 0; j < kHID; ++j)
                o += sm[OFF_H1 + j] * sm[OFF_WLIN + tid * kHID + j];
            outY[t * kOUT + tid] = o;
        }
        // (next iteration's first write is separated by barriers 1&2; safe)
    }

    if (tid < kHID) { minv[tid] = mn; maxv[tid] = mx; }
}

// =====================================================================
// Kernel 3: HPM forward + analytic VJP (cotangent = ones).
//   Z = X @ W_h1.T + b_h1 ; T = tanh(Z) ; F = -(T @ W_h2.T + b_h2)
//   dT_j = -colsum_j(W_h2) ; dZ = dT * (1 - T^2) ; AllF_x = dZ @ W_h1
// X = [outputs(10), zeros(64), input(180), minmax(old_h)(64)]
// One thread per sequence row.
// =====================================================================
__global__ __launch_bounds__(256) void hpm_vjp(
    const float* __restrict__ outY,   // [kSEQ][kOUT]
    const float* __restrict__ xin,    // [kSEQ][kIN]
    const float* __restrict__ oldh,   // [kSEQ][kHID]
    const float* __restrict__ minv,
    const float* __restrict__ maxv,
    const float* __restrict__ W1,     // [2][kHPM]
    const float* __restrict__ b1,     // [2]
    const float* __restrict__ W2,     // [kIN][2]
    const float* __restrict__ b2,     // [kIN]
    float* __restrict__ Fout,         // [kSEQ][kIN]
    float* __restrict__ Gout)         // [kSEQ][kHPM]
{
    const int s = blockIdx.x * blockDim.x + threadIdx.x;
    if (s >= kSEQ) return;

    float z0 = b1[0], z1 = b1[1];
    // segment 0: outputs, cols [0,10)
    for (int k = 0; k < kOUT; ++k) {
        const float v = outY[s * kOUT + k];
        z0 += v * W1[k];
        z1 += v * W1[kHPM + k];
    }
    // segment 1: zeros, cols [10,74) -> no contribution
    // segment 2: input, cols [74,254)
    for (int k = 0; k < kIN; ++k) {
        const float v = xin[s * kIN + k];
        z0 += v * W1[74 + k];
        z1 += v * W1[kHPM + 74 + k];
    }
    // segment 3: minmax-normalized old hidden, cols [254,318)
    for (int k = 0; k < kHID; ++k) {
        const float v = (oldh[s * kHID + k] - minv[k]) /
                        (maxv[k] - minv[k] + 1e-6f);
        z0 += v * W1[254 + k];
        z1 += v * W1[kHPM + 254 + k];
    }

    const float t0 = tanhf(z0);
    const float t1 = tanhf(z1);

    float cs0 = 0.0f, cs1 = 0.0f;
    for (int i = 0; i < kIN; ++i) {
        const float w0 = W2[2 * i];
        const float w1 = W2[2 * i + 1];
        Fout[s * kIN + i] = -(t0 * w0 + t1 * w1 + b2[i]);
        cs0 += w0;
        cs1 += w1;
    }

    const float dz0 = -cs0 * (1.0f - t0 * t0);
    const float dz1 = -cs1 * (1.0f - t1 * t1);
    for (int k = 0; k < kHPM; ++k)
        Gout[s * kHPM + k] = dz0 * W1[k] + dz1 * W1[kHPM + k];
}

// =====================================================================
extern "C" void kernel_launch(void* const* d_in, const int* in_sizes, int n_in,
                              void* d_out, int out_size, void* d_ws, size_t ws_size,
                              hipStream_t stream) {
    const float* input = (const float*)d_in[0];
    const float* Wih0  = (const float*)d_in[1];
    const float* Whh0  = (const float*)d_in[2];
    const float* bih0  = (const float*)d_in[3];
    const float* bhh0  = (const float*)d_in[4];
    const float* Wih1  = (const float*)d_in[5];
    const float* Whh1  = (const float*)d_in[6];
    const float* bih1  = (const float*)d_in[7];
    const float* bhh1  = (const float*)d_in[8];
    const float* Wlin  = (const float*)d_in[9];
    const float* blin  = (const float*)d_in[10];
    const float* Wh1   = (const float*)d_in[11];
    const float* bh1   = (const float*)d_in[12];
    const float* Wh2   = (const float*)d_in[13];
    const float* bh2   = (const float*)d_in[14];

    float* outY = (float*)d_out;                  // [8192][10]
    float* Fout = outY + kSEQ * kOUT;             // [8192][180]
    float* Gout = Fout + kSEQ * kIN;              // [8192][318]

    float* pre0 = (float*)d_ws;                   // [8192][256]
    float* oldh = pre0 + kSEQ * kG;               // [8192][64]
    float* minv = oldh + kSEQ * kHID;             // [64]
    float* maxv = minv + kHID;                    // [64]

    // 1) input projection GEMM (WMMA f32): 8192 tiles / 8 waves per block
    xproj_wmma<<<1024, 256, 0, stream>>>(input, Wih0, bih0, bhh0, pre0);

    // 2) sequential scan, one WGP, ~198KB LDS
    lstm_seq<<<1, 256, LDS_FLOATS * sizeof(float), stream>>>(
        pre0, Whh0, Wih1, Whh1, bih1, bhh1, Wlin, blin,
        outY, oldh, minv, maxv);

    // 3) HPM forward + VJP, one thread per row
    hpm_vjp<<<kSEQ / 256, 256, 0, stream>>>(
        outY, input, oldh, minv, maxv, Wh1, bh1, Wh2, bh2, Fout, Gout);
}